// RoutedNeighborhoodAttention_6846177870515
// MI455X (gfx1250) — compile-verified
//
#include <hip/hip_runtime.h>

typedef __attribute__((ext_vector_type(2))) float v2f;
typedef __attribute__((ext_vector_type(8))) float v8f;

#define D_   128
#define H_   8
#define DH_  16
#define M_   3
#define NEG_ (-1e9f)
#define XS_STRIDE 132   // floats; 528B row stride: 16B-aligned rows + conflict-free columns

// ---- order-preserving float<->uint map for integer atomicMax segment-max ----
__device__ __forceinline__ unsigned f2ord(float f) {
    unsigned u = __float_as_uint(f);
    return (u & 0x80000000u) ? ~u : (u | 0x80000000u);
}
__device__ __forceinline__ float ord2f(unsigned u) {
    return (u & 0x80000000u) ? __uint_as_float(u & 0x7fffffffu)
                             : __uint_as_float(~u);
}

// ---------------- kernel 1: init accumulators ----------------
__global__ void rna_init_kernel(float* __restrict__ out_states,
                                float* __restrict__ nsum,
                                unsigned* __restrict__ nmax, int n) {
    int i = blockIdx.x * blockDim.x + threadIdx.x;
    if (i < n * D_) out_states[i] = 0.0f;
    if (i < n * H_) { nsum[i] = 0.0f; nmax[i] = 0u; }  // 0u == "-inf" in ord space
}

// ---------------- kernel 2: fused QKV projection via fp32 WMMA ----------------
// One block = 16 rows of x; 8 waves, wave w computes output columns [16w,16w+16)
// for Q, K and V simultaneously (x tile reused from LDS).
__global__ __launch_bounds__(256) void rna_qkv_wmma_kernel(
    const float* __restrict__ x,
    const float* __restrict__ Wq, const float* __restrict__ bq,
    const float* __restrict__ Wk, const float* __restrict__ bk,
    const float* __restrict__ Wv, const float* __restrict__ bv,
    float* __restrict__ q, float* __restrict__ k, float* __restrict__ v,
    int n)
{
    __shared__ float xs[16][XS_STRIDE];
    const int tid  = threadIdx.x;
    const int row0 = blockIdx.x * 16;

    // vectorized tile fill: 512 float4 groups, 2 per thread
#pragma unroll
    for (int it = 0; it < 2; ++it) {
        const int g  = tid + 256 * it;       // group id
        const int r  = g >> 5;               // 32 groups (128 floats) per row
        const int c4 = (g & 31) << 2;
        const int gr = row0 + r;
        float4 val = make_float4(0.f, 0.f, 0.f, 0.f);
        if (gr < n)
            val = *(const float4*)(x + (size_t)gr * D_ + c4);
        *(float4*)&xs[r][c4] = val;          // ds_store_b128, rows 16B-aligned
    }
    __syncthreads();

    const int wave = tid >> 5;
    const int lane = tid & 31;
    const int half = lane >> 4;          // K-pair / M-half selector
    const int l16  = lane & 15;
    const int col  = wave * 16 + l16;    // output column owned by this lane (B/C/D)

    v8f cq = {}; v8f ck = {}; v8f cv = {};
    for (int kk = 0; kk < D_; kk += 4) {
        const int ka = kk + 2 * half;
        // A (16x4, M x K): lanes 0-15 hold K=kk,kk+1; lanes 16-31 hold K=kk+2,kk+3
        v2f a = *(const v2f*)&xs[l16][ka];            // single ds_load_b64
        // B (4x16, K x N): mirrored striping, lane indexes N
        v2f bqf, bkf, bvf;
        bqf.x = Wq[(size_t)ka * D_ + col]; bqf.y = Wq[(size_t)(ka + 1) * D_ + col];
        bkf.x = Wk[(size_t)ka * D_ + col]; bkf.y = Wk[(size_t)(ka + 1) * D_ + col];
        bvf.x = Wv[(size_t)ka * D_ + col]; bvf.y = Wv[(size_t)(ka + 1) * D_ + col];
        cq = __builtin_amdgcn_wmma_f32_16x16x4_f32(false, a, false, bqf, (short)0, cq, false, false);
        ck = __builtin_amdgcn_wmma_f32_16x16x4_f32(false, a, false, bkf, (short)0, ck, false, false);
        cv = __builtin_amdgcn_wmma_f32_16x16x4_f32(false, a, false, bvf, (short)0, cv, false, false);
    }

    const float biasq = bq[col], biask = bk[col], biasv = bv[col];
#pragma unroll
    for (int i = 0; i < 8; ++i) {
        const int m  = i + 8 * half;     // C/D layout: VGPR i -> M = i (+8 for upper half)
        const int gr = row0 + m;
        if (gr < n) {
            q[(size_t)gr * D_ + col] = cq[i] + biasq;
            k[(size_t)gr * D_ + col] = ck[i] + biask;
            v[(size_t)gr * D_ + col] = cv[i] + biasv;
        }
    }
}

// ---------------- kernel 3: per-(edge,head) masked score + segment max ----------------
__global__ __launch_bounds__(256) void rna_score_kernel(
    const float* __restrict__ q, const float* __restrict__ k,
    const float* __restrict__ pi, const float* __restrict__ vw,
    const int* __restrict__ src, const int* __restrict__ dst,
    float* __restrict__ scores, unsigned* __restrict__ nmax, int eh)
{
    int idx = blockIdx.x * blockDim.x + threadIdx.x;
    if (idx >= eh) return;
    const int e = idx >> 3, h = idx & 7;
    const int d = dst[e], s = src[e];

    const float4* q4 = (const float4*)(q + (size_t)d * D_ + h * DH_);
    const float4* k4 = (const float4*)(k + (size_t)s * D_ + h * DH_);
    float dot = 0.0f;
#pragma unroll
    for (int i = 0; i < 4; ++i) {
        float4 a = q4[i], b = k4[i];
        dot += a.x * b.x + a.y * b.y + a.z * b.z + a.w * b.w;
    }

    float mix = 0.0f;
#pragma unroll
    for (int m = 0; m < M_; ++m)
        mix += pi[((size_t)d * H_ + h) * M_ + m] * vw[(size_t)e * M_ + m];

    const float sc = (mix > 0.0f) ? dot * 0.25f + logf(fmaxf(mix, 1e-8f)) : NEG_;
    scores[idx] = sc;
    atomicMax(&nmax[(size_t)d * H_ + h], f2ord(sc));
}

// ---------------- kernel 4: expv + segment sum ----------------
__global__ __launch_bounds__(256) void rna_exp_kernel(
    const float* __restrict__ scores, const unsigned* __restrict__ nmax,
    const int* __restrict__ dst,
    float* __restrict__ attn, float* __restrict__ nsum, int eh)
{
    int idx = blockIdx.x * blockDim.x + threadIdx.x;
    if (idx >= eh) return;
    const int e = idx >> 3, h = idx & 7;
    const int d = dst[e];
    const float sc = scores[idx];
    float ev = 0.0f;
    if (sc > -1e8f) {                               // valid edges only
        const float smax = ord2f(nmax[(size_t)d * H_ + h]);
        ev = expf(sc - smax);
    }
    attn[idx] = ev;                                  // unnormalized for now
    if (ev != 0.0f) atomicAdd(&nsum[(size_t)d * H_ + h], ev);
}

// ---------------- kernel 5: normalize + weighted scatter-sum of v ----------------
__global__ __launch_bounds__(256) void rna_agg_kernel(
    float* __restrict__ attn, const float* __restrict__ nsum,
    const float* __restrict__ v,
    const int* __restrict__ src, const int* __restrict__ dst,
    float* __restrict__ out_states, int eh)
{
    int idx = blockIdx.x * blockDim.x + threadIdx.x;
    if (idx >= eh) return;
    const int e = idx >> 3, h = idx & 7;
    const int d = dst[e], s = src[e];
    const float ssum = nsum[(size_t)d * H_ + h];
    const float a = attn[idx] / fmaxf(ssum, 1e-8f);
    attn[idx] = a;
    if (a != 0.0f) {
        const float* vp = v + (size_t)s * D_ + h * DH_;
        float* op = out_states + (size_t)d * D_ + h * DH_;
#pragma unroll
        for (int j = 0; j < DH_; ++j)
            atomicAdd(&op[j], vp[j] * a);            // L2-resident fp32 atomics
    }
}

extern "C" void kernel_launch(void* const* d_in, const int* in_sizes, int n_in,
                              void* d_out, int out_size, void* d_ws, size_t ws_size,
                              hipStream_t stream) {
    const float* x  = (const float*)d_in[0];
    const float* pi = (const float*)d_in[1];
    const float* vw = (const float*)d_in[2];
    const float* Wq = (const float*)d_in[3];
    const float* bq = (const float*)d_in[4];
    const float* Wk = (const float*)d_in[5];
    const float* bk = (const float*)d_in[6];
    const float* Wv = (const float*)d_in[7];
    const float* bv = (const float*)d_in[8];
    const int* src  = (const int*)d_in[9];
    const int* dst  = (const int*)d_in[10];

    const int N = in_sizes[0] / D_;
    const int E = in_sizes[9];

    float* out  = (float*)d_out;                    // node_states [N*128]
    float* attn = out + (size_t)N * D_;             // attn [E*8]

    // scratch layout
    float*    q      = (float*)d_ws;
    float*    kbuf   = q    + (size_t)N * D_;
    float*    vbuf   = kbuf + (size_t)N * D_;
    float*    scores = vbuf + (size_t)N * D_;
    unsigned* nmax   = (unsigned*)(scores + (size_t)E * H_);
    float*    nsum   = (float*)(nmax + (size_t)N * H_);

    const int initN = N * D_;
    rna_init_kernel<<<(initN + 255) / 256, 256, 0, stream>>>(out, nsum, nmax, N);

    rna_qkv_wmma_kernel<<<(N + 15) / 16, 256, 0, stream>>>(
        x, Wq, bq, Wk, bk, Wv, bv, q, kbuf, vbuf, N);

    const int EH  = E * H_;
    const int blk = (EH + 255) / 256;
    rna_score_kernel<<<blk, 256, 0, stream>>>(q, kbuf, pi, vw, src, dst, scores, nmax, EH);
    rna_exp_kernel  <<<blk, 256, 0, stream>>>(scores, nmax, dst, attn, nsum, EH);
    rna_agg_kernel  <<<blk, 256, 0, stream>>>(attn, nsum, vbuf, src, dst, out, EH);
}